// WeightedLabelSmoothCELoss_68633577390300
// MI455X (gfx1250) — compile-verified
//
#include <hip/hip_runtime.h>

typedef float v2f __attribute__((ext_vector_type(2)));
typedef float v4f __attribute__((ext_vector_type(4)));
typedef int   v4i __attribute__((ext_vector_type(4)));
typedef float v8f __attribute__((ext_vector_type(8)));

#define NBLOCKS  2048
#define NTHREADS 256

// w * (lse - 0.9*p[label] - (1/30)*(p0+p1+p2))   [weights: 10, 10/3, 5/3]
__device__ __forceinline__ float weighted_sample_loss(float p0, float p1, float p2, int lab) {
    const float L2E = 1.4426950408889634f;   // log2(e)
    const float LN2 = 0.6931471805599453f;   // ln(2)
    float m   = fmaxf(fmaxf(p0, p1), p2);                        // v_max3_f32
    float t   = __builtin_amdgcn_exp2f((p0 - m) * L2E)
              + __builtin_amdgcn_exp2f((p1 - m) * L2E)
              + __builtin_amdgcn_exp2f((p2 - m) * L2E);          // 3x v_exp_f32
    float lse = fmaf(__builtin_amdgcn_logf(t), LN2, m);          // v_log_f32 + fma
    float psum = p0 + p1 + p2;
    float pl  = (lab == 0) ? p0 : ((lab == 1) ? p1 : p2);
    float w   = (lab == 0) ? 10.0f : ((lab == 1) ? (10.0f / 3.0f) : (5.0f / 3.0f));
    return w * (lse - 0.9f * pl - psum * (1.0f / 30.0f));
}

// Streaming kernel: 4 samples / thread / iter -> 3x b128 pred loads + 1x b128 label load.
__global__ void __launch_bounds__(NTHREADS)
ce_partial_kernel(const float* __restrict__ pred, const int* __restrict__ label,
                  float* __restrict__ partials, int nChunks, int nSamples) {
    const int tid    = blockIdx.x * NTHREADS + threadIdx.x;
    const int stride = gridDim.x * NTHREADS;

    float acc = 0.0f;
    for (int c = tid; c < nChunks; c += stride) {
        const v4f* p  = (const v4f*)(pred + (size_t)c * 12);
        const v4i* lb = (const v4i*)(label + (size_t)c * 4);
        v4f f0 = p[0];
        v4f f1 = p[1];
        v4f f2 = p[2];
        v4i l  = lb[0];
        acc += weighted_sample_loss(f0[0], f0[1], f0[2], l[0]);
        acc += weighted_sample_loss(f0[3], f1[0], f1[1], l[1]);
        acc += weighted_sample_loss(f1[2], f1[3], f2[0], l[2]);
        acc += weighted_sample_loss(f2[1], f2[2], f2[3], l[3]);
    }
    // Tail samples (none for N divisible by 4, kept for generality)
    for (int s = nChunks * 4 + tid; s < nSamples; s += stride) {
        const float* ps = pred + (size_t)s * 3;
        acc += weighted_sample_loss(ps[0], ps[1], ps[2], label[s]);
    }

    // wave32 shuffle reduction (ds_bpermute path)
    for (int off = 16; off > 0; off >>= 1)
        acc += __shfl_down(acc, off, 32);

    __shared__ float wsum[NTHREADS / 32];
    const int lane = threadIdx.x & 31;
    const int wid  = threadIdx.x >> 5;
    if (lane == 0) wsum[wid] = acc;
    __syncthreads();
    if (threadIdx.x == 0) {
        float s = 0.0f;
#pragma unroll
        for (int i = 0; i < NTHREADS / 32; ++i) s += wsum[i];
        partials[blockIdx.x] = s;
    }
}

// Final reduction, single wave32. Last 32->1 step uses V_WMMA_F32_16X16X4_F32 as a
// ones-matvec reduction: A(16x4) holds the 32 lane partials (K0 in lanes 0-15,
// K2 in lanes 16-31, K1/K3 zero), B = ones(4x16) -> D row m = partial[m] + partial[m+16],
// replicated across columns. Sum of D's 8 VGPRs in lane 0 = rows 0..7, lane 16 = rows 8..15.
__global__ void ce_final_kernel(const float* __restrict__ partials, int n,
                                float* __restrict__ out, float invN) {
    const int lane = threadIdx.x;   // 0..31, launched with exactly 32 threads
    float acc = 0.0f;
    for (int i = lane; i < n; i += 32) acc += partials[i];

    v2f a; a[0] = acc;  a[1] = 0.0f;   // A matrix: 2 VGPRs (16x4 f32)
    v2f b; b[0] = 1.0f; b[1] = 1.0f;   // B matrix: 2 VGPRs, all ones (4x16 f32)
    v8f c = {};
    v8f d = __builtin_amdgcn_wmma_f32_16x16x4_f32(
        /*neg_a=*/false, a, /*neg_b=*/false, b,
        /*c_mod=*/(short)0, c, /*reuse_a=*/false, /*reuse_b=*/false);

    float rs = d[0] + d[1] + d[2] + d[3] + d[4] + d[5] + d[6] + d[7];

    __shared__ float sm[2];
    if (lane == 0)  sm[0] = rs;   // sum of rows 0..7  = partials of lanes 0..7  + 16..23
    if (lane == 16) sm[1] = rs;   // sum of rows 8..15
    __syncthreads();
    if (lane == 0) out[0] = (sm[0] + sm[1]) * invN;
}

extern "C" void kernel_launch(void* const* d_in, const int* in_sizes, int n_in,
                              void* d_out, int out_size, void* d_ws, size_t ws_size,
                              hipStream_t stream) {
    const float* pred  = (const float*)d_in[0];
    const int*   label = (const int*)d_in[1];
    float*       out   = (float*)d_out;

    const int N       = in_sizes[1];   // number of samples (label count)
    const int nChunks = N / 4;

    int nblocks = NBLOCKS;
    if (ws_size < (size_t)nblocks * sizeof(float))
        nblocks = (int)(ws_size / sizeof(float));   // safety clamp
    float* partials = (float*)d_ws;

    ce_partial_kernel<<<nblocks, NTHREADS, 0, stream>>>(pred, label, partials, nChunks, N);
    ce_final_kernel<<<1, 32, 0, stream>>>(partials, nblocks, out, 1.0f / (float)N);
}